// GCNEncoder_52596169507006
// MI455X (gfx1250) — compile-verified
//
#include <hip/hip_runtime.h>
#include <hip/hip_bf16.h>

typedef float v2f __attribute__((ext_vector_type(2)));
typedef float v8f __attribute__((ext_vector_type(8)));
typedef float v4f __attribute__((ext_vector_type(4)));

#define N_NODES 65536
#define N_EDGES 1048576

// ---------------------------------------------------------------- utilities
__global__ void zero_f32(float* __restrict__ p, size_t n) {
    size_t i = (size_t)blockIdx.x * blockDim.x + threadIdx.x;
    size_t stride = (size_t)gridDim.x * blockDim.x;
    for (; i < n; i += stride) p[i] = 0.0f;
}

// deg[c] += w[e]  (self-loop +1 folded into rsqrt kernel)
__global__ void deg_kernel(const int* __restrict__ eidx, const float* __restrict__ ew,
                           float* __restrict__ deg, int E) {
    int e = blockIdx.x * blockDim.x + threadIdx.x;
    if (e < E) {
        int c = eidx[E + e];           // edge_index[1][e]
        atomicAdd(&deg[c], ew[e]);
    }
}

// dis[n] = rsqrt(deg[n] + 1)   (deg_total >= 1 > 0 always)
__global__ void rsqrt1p_kernel(float* __restrict__ d, int n) {
    int i = blockIdx.x * blockDim.x + threadIdx.x;
    if (i < n) d[i] = rsqrtf(d[i] + 1.0f);
}

// norm[e] = dis[row]*w*dis[col]
__global__ void norm_kernel(const int* __restrict__ eidx, const float* __restrict__ ew,
                            const float* __restrict__ dis, float* __restrict__ norm, int E) {
    int e = blockIdx.x * blockDim.x + threadIdx.x;
    if (e < E) {
        int r = eidx[e];
        int c = eidx[N_EDGES + e];
        norm[e] = dis[r] * ew[e] * dis[c];
    }
}

// ------------------------------------------------------------- aggregation
// One wave per edge; each lane handles WIDTH/32 consecutive floats.
template<int WIDTH, bool PAD>
__global__ void scatter_edges(const float* __restrict__ h, const int* __restrict__ eidx,
                              const float* __restrict__ norm, float* __restrict__ out, int E) {
    int gid  = blockIdx.x * blockDim.x + threadIdx.x;
    int e    = gid >> 5;
    int lane = gid & 31;
    if (e >= E) return;
    int r = eidx[e];
    int c = eidx[N_EDGES + e];
    float nw = norm[e];
    constexpr int PL = WIDTH / 32;                     // 4 (w=128) or 2 (w=64)
    const float* src = h + (size_t)r * WIDTH + lane * PL;
    size_t dbase;
    if (PAD) dbase = (size_t)c * 64 + (size_t)(c >> 10) * 64 + lane * PL;  // padded (64,1025,64)
    else     dbase = (size_t)c * WIDTH + lane * PL;
    float v[PL];
#pragma unroll
    for (int i = 0; i < PL; ++i) v[i] = src[i];
#pragma unroll
    for (int i = 0; i < PL; ++i) atomicAdd(&out[dbase + i], v[i] * nw);
}

// Per-element: out += h[n]*dis[n]^2 (+bias) (relu); no atomics needed (post-scatter pass).
template<int WIDTH, bool RELU, bool PAD>
__global__ void finalize_nodes(const float* __restrict__ h, const float* __restrict__ dis,
                               const float* __restrict__ bias, float* __restrict__ out, int n_nodes) {
    int gid = blockIdx.x * blockDim.x + threadIdx.x;
    int n = gid / WIDTH;
    int c = gid % WIDTH;
    if (n >= n_nodes) return;
    size_t idx;
    if (PAD) idx = (size_t)n * 64 + (size_t)(n >> 10) * 64 + c;
    else     idx = (size_t)n * WIDTH + c;
    float d  = dis[n];
    float v  = out[idx] + h[(size_t)n * WIDTH + c] * d * d;
    if (bias) v += bias[c];
    if (RELU) v = fmaxf(v, 0.0f);
    out[idx] = v;
}

// ------------------------------------------------------------- WMMA GEMM
// OUT[M,N] = A[M,K] * W[K,N] (+bias, relu). Pure fp32 via V_WMMA_F32_16X16X4_F32.
// Block = 128 threads = 4 waves; each wave owns one 16-row tile and a 16x64 strip.
// Lane layout per ISA: lanes 0-15 carry K={k,k+1}/M=0..15, lanes 16-31 carry K={k+2,k+3}/M=0..15.
template<bool BIAS_RELU>
__global__ void gemm_wmma_f32(const float* __restrict__ A, const float* __restrict__ W,
                              const float* __restrict__ bias, float* __restrict__ OUT,
                              int K, int N) {
    const int lane = threadIdx.x & 31;
    const int wave = threadIdx.x >> 5;
    const int half = lane >> 4;       // 0: lanes 0-15, 1: lanes 16-31
    const int r    = lane & 15;
    const int m0   = (blockIdx.x * 4 + wave) * 16;
    const int n0   = blockIdx.y * 64;

    v8f acc[4] = {};

    const float* Arow = A + (size_t)(m0 + r) * K + 2 * half;
    for (int k = 0; k < K; k += 4) {
        v2f a;
        a.x = Arow[k + 0];
        a.y = Arow[k + 1];
        const float* W0 = W + (size_t)(k + 2 * half) * N;
#pragma unroll
        for (int t = 0; t < 4; ++t) {
            int nb = n0 + t * 16 + r;
            v2f b;
            b.x = W0[nb];
            b.y = W0[N + nb];
            acc[t] = __builtin_amdgcn_wmma_f32_16x16x4_f32(
                false, a, false, b, (short)0, acc[t], false, false);
        }
    }

#pragma unroll
    for (int t = 0; t < 4; ++t) {
        int nb = n0 + t * 16 + r;
        float bv = BIAS_RELU ? bias[nb] : 0.0f;
#pragma unroll
        for (int i = 0; i < 8; ++i) {
            int row = m0 + i + 8 * half;      // C/D layout: VGPR i -> M=i (lo half) / M=i+8 (hi half)
            float v = acc[t][i];
            if (BIAS_RELU) v = fmaxf(v + bv, 0.0f);
            OUT[(size_t)row * N + nb] = v;
        }
    }
}

// ---------------------------------------------------------------- driver
extern "C" void kernel_launch(void* const* d_in, const int* in_sizes, int n_in,
                              void* d_out, int out_size, void* d_ws, size_t ws_size,
                              hipStream_t stream) {
    const float* x  = (const float*)d_in[0];
    const int*   ei = (const int*)d_in[1];
    const float* ew = (const float*)d_in[2];
    const float* W1 = (const float*)d_in[4];
    const float* b1 = (const float*)d_in[5];
    const float* W2 = (const float*)d_in[6];
    const float* b2 = (const float*)d_in[7];
    const float* W3 = (const float*)d_in[8];
    const float* b3 = (const float*)d_in[9];
    float* out = (float*)d_out;

    const int N = N_NODES, E = N_EDGES;

    // workspace carve-out
    char* ws = (char*)d_ws;
    float* norm = (float*)ws;  ws += (size_t)E * 4;            //  4 MB
    float* dis  = (float*)ws;  ws += (size_t)N * 4;            //  256 KB (deg, then dis in place)
    float* B0   = (float*)ws;  ws += (size_t)N * 128 * 4;      // 32 MB : agg0, later h2
    float* B1   = (float*)ws;  ws += (size_t)N * 256 * 4;      // 64 MB : h1
    float* B2   = (float*)ws;                                   // 32 MB : t2, later t3

    const int TB = 256;
    const int egrid  = E / TB;             // elementwise over edges
    const int sgrid  = (E * 32) / TB;      // wave-per-edge scatter
    dim3 gblk(128);

    // 1) degree + dis + norm
    zero_f32<<<512, TB, 0, stream>>>(dis, (size_t)N);
    deg_kernel<<<egrid, TB, 0, stream>>>(ei, ew, dis, E);
    rsqrt1p_kernel<<<N / TB, TB, 0, stream>>>(dis, N);
    norm_kernel<<<egrid, TB, 0, stream>>>(ei, ew, dis, norm, E);

    // 2) layer 1: agg first (width 128), then GEMM(+b1,relu) to 256
    zero_f32<<<4096, TB, 0, stream>>>(B0, (size_t)N * 128);
    scatter_edges<128, false><<<sgrid, TB, 0, stream>>>(x, ei, norm, B0, E);
    finalize_nodes<128, false, false><<<(N * 128) / TB, TB, 0, stream>>>(x, dis, nullptr, B0, N);
    gemm_wmma_f32<true><<<dim3(1024, 4), gblk, 0, stream>>>(B0, W1, b1, B1, 128, 256);

    // 3) layer 2: GEMM first (to 128), then agg + b2 + relu
    gemm_wmma_f32<false><<<dim3(1024, 2), gblk, 0, stream>>>(B1, W2, nullptr, B2, 256, 128);
    zero_f32<<<4096, TB, 0, stream>>>(B0, (size_t)N * 128);
    scatter_edges<128, false><<<sgrid, TB, 0, stream>>>(B2, ei, norm, B0, E);
    finalize_nodes<128, true, false><<<(N * 128) / TB, TB, 0, stream>>>(B2, dis, b2, B0, N);

    // 4) layer 3: GEMM first (to 64), then agg + b3 straight into padded output
    gemm_wmma_f32<false><<<dim3(1024, 1), gblk, 0, stream>>>(B0, W3, nullptr, B2, 128, 64);
    zero_f32<<<4096, TB, 0, stream>>>(out, (size_t)out_size);   // also creates the pad rows
    scatter_edges<64, true><<<sgrid, TB, 0, stream>>>(B2, ei, norm, out, E);
    finalize_nodes<64, false, true><<<(N * 64) / TB, TB, 0, stream>>>(B2, dis, b3, out, N);
}